// GAT_13589276524899
// MI455X (gfx1250) — compile-verified
//
#include <hip/hip_runtime.h>
#include <math.h>

// ---------------- constants from the reference ----------------
#define NNODES    20000
#define NEDGES    320000
#define NODE_DIM  256
#define HID       64
#define HEADS     4
#define OUTC      50
#define NEG_SLOPE 0.2f
#define BN_EPS    1e-5f

typedef float v2f __attribute__((ext_vector_type(2)));
typedef float v8f __attribute__((ext_vector_type(8)));

// ---------------- fp32 WMMA GEMM: C[M,N] = A[M,K] @ B[K,N] ----------------
// One wave per 16x16 C tile; K stepped by 4 through v_wmma_f32_16x16x4_f32.
// Requires M%16==0, N%16==0, K%4==0 (true for all calls here) so EXEC is all-1s.
__global__ void __launch_bounds__(256)
wmma_gemm_f32(const float* __restrict__ A, const float* __restrict__ B,
              float* __restrict__ C, int M, int N, int K) {
    int wave = (int)((blockIdx.x * blockDim.x + threadIdx.x) >> 5);
    int lane = threadIdx.x & 31;
    int tilesN = N >> 4;
    int totalTiles = (M >> 4) * tilesN;
    if (wave >= totalTiles) return;                 // wave-uniform exit

    int tm   = (wave / tilesN) << 4;
    int tn   = (wave % tilesN) << 4;
    int half = lane >> 4;                           // 0: lanes 0-15, 1: lanes 16-31
    int l16  = lane & 15;

    const float* __restrict__ arow = A + (size_t)(tm + l16) * K + (half << 1);
    const float* __restrict__ bcol = B + (size_t)(half << 1) * N + (tn + l16);

    v8f acc = {};
    for (int k = 0; k < K; k += 4) {
        v2f a, b;
        a[0] = arow[k];
        a[1] = arow[k + 1];
        b[0] = bcol[(size_t)k * N];
        b[1] = bcol[(size_t)(k + 1) * N];
        acc = __builtin_amdgcn_wmma_f32_16x16x4_f32(
            /*neg_a=*/false, a, /*neg_b=*/false, b,
            /*c_mod=*/(short)0, acc, /*reuse_a=*/false, /*reuse_b=*/false);
    }

    float* __restrict__ crow = C + (size_t)(tm + (half << 3)) * N + tn + l16;
#pragma unroll
    for (int i = 0; i < 8; ++i) crow[(size_t)i * N] = acc[i];
}

// ---------------- small utility kernels ----------------
__global__ void fill_f32(float* p, float v, size_t n) {
    size_t i = blockIdx.x * (size_t)blockDim.x + threadIdx.x;
    if (i < n) p[i] = v;
}

__global__ void recip_eps(float* p, size_t n) {
    size_t i = blockIdx.x * (size_t)blockDim.x + threadIdx.x;
    if (i < n) p[i] = 1.0f / (p[i] + 1e-16f);
}

// monotone float atomic-max via int/uint encodings (init must be -inf)
__device__ __forceinline__ void atomic_max_f32(float* addr, float v) {
    if (v >= 0.0f) atomicMax((int*)addr, __float_as_int(v));
    else           atomicMin((unsigned int*)addr, __float_as_uint(v));
}

// ---------------- edge pass 1: attention scores + segment max ----------------
// thread t -> (edge e = t/H, head h = t%H); dot over D dims of leaky_relu(xi+xj)
__global__ void edge_scores(const float* __restrict__ xl, const float* __restrict__ xr,
                            const int* __restrict__ src, const int* __restrict__ dst,
                            const float* __restrict__ att,
                            float* __restrict__ score, float* __restrict__ nmax,
                            int E, int H, int D) {
    int t = blockIdx.x * blockDim.x + threadIdx.x;
    if (t >= E * H) return;
    int e = t / H, h = t - e * H;
    int s = src[e], d = dst[e];
    const float* pj = xl + (size_t)s * H * D + (size_t)h * D;
    const float* pi = xr + (size_t)d * H * D + (size_t)h * D;
    const float* pa = att + (size_t)h * D;
    float acc = 0.0f;
    for (int k = 0; k < D; ++k) {
        float v = pi[k] + pj[k];
        v = (v > 0.0f) ? v : NEG_SLOPE * v;
        acc = fmaf(v, pa[k], acc);
    }
    score[t] = acc;
    atomic_max_f32(&nmax[(size_t)d * H + h], acc);
}

// ---------------- edge pass 2: p = exp(score - max[dst]); denom += p ----------------
__global__ void edge_exp(float* __restrict__ score, const float* __restrict__ nmax,
                         float* __restrict__ den, const int* __restrict__ dst,
                         int E, int H) {
    int t = blockIdx.x * blockDim.x + threadIdx.x;
    if (t >= E * H) return;
    int e = t / H, h = t - e * H;
    int d = dst[e];
    float p = __expf(score[t] - nmax[(size_t)d * H + h]);
    score[t] = p;
    atomicAdd(&den[(size_t)d * H + h], p);
}

// ---------------- edge pass 3: out[dst] += alpha * xl[src] ----------------
// thread t -> (edge e = t/C, channel c = t%C); consecutive threads hit
// consecutive channels of one destination row -> coalesced atomics.
__global__ void edge_agg(const float* __restrict__ p, const float* __restrict__ invden,
                         const float* __restrict__ xl,
                         const int* __restrict__ src, const int* __restrict__ dst,
                         float* __restrict__ out, int E, int H, int D, int C) {
    size_t t = blockIdx.x * (size_t)blockDim.x + threadIdx.x;
    if (t >= (size_t)E * C) return;
    int e = (int)(t / C);
    int c = (int)(t - (size_t)e * C);
    int h = c / D;
    int s = src[e], d = dst[e];
    float alpha = p[(size_t)e * H + h] * invden[(size_t)d * H + h];
    atomicAdd(&out[(size_t)d * C + c], alpha * xl[(size_t)s * C + c]);
}

// ---------------- batchnorm statistics: one block per channel ----------------
__global__ void bn_stats(const float* __restrict__ x, int rows, int C,
                         float* __restrict__ mu, float* __restrict__ rs) {
    __shared__ float sh[256], sh2[256];
    int c = blockIdx.x, t = threadIdx.x;
    float s = 0.0f, s2 = 0.0f;
    for (int r = t; r < rows; r += 256) {
        float v = x[(size_t)r * C + c];
        s += v; s2 = fmaf(v, v, s2);
    }
    sh[t] = s; sh2[t] = s2;
    __syncthreads();
    for (int off = 128; off > 0; off >>= 1) {
        if (t < off) { sh[t] += sh[t + off]; sh2[t] += sh2[t + off]; }
        __syncthreads();
    }
    if (t == 0) {
        float m   = sh[0] / (float)rows;
        float var = sh2[0] / (float)rows - m * m;
        mu[c] = m;
        rs[c] = rsqrtf(var + BN_EPS);
    }
}

// ---------------- BN apply + exact GELU (in place) ----------------
__global__ void bn_gelu(float* __restrict__ x, const float* __restrict__ mu,
                        const float* __restrict__ rs, const float* __restrict__ g,
                        const float* __restrict__ b, size_t total, int C) {
    size_t i = blockIdx.x * (size_t)blockDim.x + threadIdx.x;
    if (i >= total) return;
    int c = (int)(i % C);
    float v = (x[i] - mu[c]) * rs[c] * g[c] + b[c];
    x[i] = 0.5f * v * (1.0f + erff(v * 0.70710678118654752f));
}

// ---------------- classifier + log_softmax: one 64-thread block per row ----------------
__global__ void __launch_bounds__(64)
classify_logsoftmax(const float* __restrict__ h, const float* __restrict__ Wc,
                    const float* __restrict__ bc, float* __restrict__ out, int rows) {
    __shared__ float row[HID];
    __shared__ float logit[64];
    __shared__ float red[2];
    int r = blockIdx.x, t = threadIdx.x;
    if (r >= rows) return;
    row[t] = h[(size_t)r * HID + t];
    __syncthreads();
    if (t < OUTC) {
        float acc = bc[t];
        for (int k = 0; k < HID; ++k) acc = fmaf(row[k], Wc[(size_t)k * OUTC + t], acc);
        logit[t] = acc;
    }
    __syncthreads();
    if (t == 0) {
        float mx = -INFINITY;
        for (int i = 0; i < OUTC; ++i) mx = fmaxf(mx, logit[i]);
        float s = 0.0f;
        for (int i = 0; i < OUTC; ++i) s += __expf(logit[i] - mx);
        red[0] = mx; red[1] = logf(s);
    }
    __syncthreads();
    if (t < OUTC) out[(size_t)r * OUTC + t] = logit[t] - red[0] - red[1];
}

// ---------------- host side ----------------
static inline int cdiv(long long a, long long b) { return (int)((a + b - 1) / b); }

extern "C" void kernel_launch(void* const* d_in, const int* in_sizes, int n_in,
                              void* d_out, int out_size, void* d_ws, size_t ws_size,
                              hipStream_t stream) {
    // inputs per setup_inputs() order
    const float* x     = (const float*)d_in[0];
    const int*   eidx  = (const int*)d_in[1];     // [2, E] int32 (JAX default x64-off)
    const float* W1l   = (const float*)d_in[2];
    const float* W1r   = (const float*)d_in[3];
    const float* att1  = (const float*)d_in[4];
    // d_in[5] = b1 (cancels through BN mean-subtraction)
    const float* bn1g  = (const float*)d_in[6];
    const float* bn1b  = (const float*)d_in[7];
    const float* W2l   = (const float*)d_in[8];
    const float* W2r   = (const float*)d_in[9];
    const float* att2  = (const float*)d_in[10];
    // d_in[11] = b2 (cancels through BN)
    const float* bn2g  = (const float*)d_in[12];
    const float* bn2b  = (const float*)d_in[13];
    const float* Wc    = (const float*)d_in[14];
    const float* bc    = (const float*)d_in[15];
    float*       out   = (float*)d_out;

    const int N = in_sizes[0] / NODE_DIM;   // 20000
    const int E = in_sizes[1] / 2;          // 320000
    const int C1 = HEADS * HID;             // 256
    const int C2 = HID;                     // 64
    const int* src = eidx;
    const int* dst = eidx + E;

    // workspace arena (floats)
    float* w = (float*)d_ws;
    size_t o = 0;
    float* xl1  = w + o; o += (size_t)N * C1;
    float* xr1  = w + o; o += (size_t)N * C1;
    float* h1   = w + o; o += (size_t)N * C1;
    float* xl2  = w + o; o += (size_t)N * C2;
    float* xr2  = w + o; o += (size_t)N * C2;
    float* h2   = w + o; o += (size_t)N * C2;
    float* sc1  = w + o; o += (size_t)E * HEADS;   // scores -> p (in place)
    float* sc2  = w + o; o += (size_t)E;
    float* mx1  = w + o; o += (size_t)N * HEADS;
    float* dn1  = w + o; o += (size_t)N * HEADS;
    float* mx2  = w + o; o += (size_t)N;
    float* dn2  = w + o; o += (size_t)N;
    float* mu1  = w + o; o += C1;
    float* rs1  = w + o; o += C1;
    float* mu2  = w + o; o += C2;
    float* rs2  = w + o; o += C2;
    (void)ws_size; (void)n_in; (void)out_size;

    // init accumulators every call (deterministic)
    hipMemsetAsync(h1,  0, (size_t)N * C1 * sizeof(float), stream);
    hipMemsetAsync(h2,  0, (size_t)N * C2 * sizeof(float), stream);
    hipMemsetAsync(dn1, 0, (size_t)N * HEADS * sizeof(float), stream);
    hipMemsetAsync(dn2, 0, (size_t)N * sizeof(float), stream);
    fill_f32<<<cdiv((size_t)N * HEADS, 256), 256, 0, stream>>>(mx1, -INFINITY, (size_t)N * HEADS);
    fill_f32<<<cdiv(N, 256), 256, 0, stream>>>(mx2, -INFINITY, (size_t)N);

    // ---- layer 1: dual GEMM (WMMA f32), M=20000 K=256 N=256 ----
    {
        int waves = (N >> 4) * (C1 >> 4);
        int blocks = cdiv((long long)waves * 32, 256);
        wmma_gemm_f32<<<blocks, 256, 0, stream>>>(x, W1l, xl1, N, C1, NODE_DIM);
        wmma_gemm_f32<<<blocks, 256, 0, stream>>>(x, W1r, xr1, N, C1, NODE_DIM);
    }
    // ---- layer 1: edge softmax aggregation ----
    edge_scores<<<cdiv((long long)E * HEADS, 256), 256, 0, stream>>>(
        xl1, xr1, src, dst, att1, sc1, mx1, E, HEADS, HID);
    edge_exp<<<cdiv((long long)E * HEADS, 256), 256, 0, stream>>>(
        sc1, mx1, dn1, dst, E, HEADS);
    recip_eps<<<cdiv((long long)N * HEADS, 256), 256, 0, stream>>>(dn1, (size_t)N * HEADS);
    edge_agg<<<cdiv((long long)E * C1, 256), 256, 0, stream>>>(
        sc1, dn1, xl1, src, dst, h1, E, HEADS, HID, C1);
    // ---- BN1 + GELU ----
    bn_stats<<<C1, 256, 0, stream>>>(h1, N, C1, mu1, rs1);
    bn_gelu<<<cdiv((long long)N * C1, 256), 256, 0, stream>>>(
        h1, mu1, rs1, bn1g, bn1b, (size_t)N * C1, C1);

    // ---- layer 2: dual GEMM (WMMA f32), M=20000 K=256 N=64 ----
    {
        int waves = (N >> 4) * (C2 >> 4);
        int blocks = cdiv((long long)waves * 32, 256);
        wmma_gemm_f32<<<blocks, 256, 0, stream>>>(h1, W2l, xl2, N, C2, C1);
        wmma_gemm_f32<<<blocks, 256, 0, stream>>>(h1, W2r, xr2, N, C2, C1);
    }
    // ---- layer 2: edge softmax aggregation (H=1) ----
    edge_scores<<<cdiv(E, 256), 256, 0, stream>>>(
        xl2, xr2, src, dst, att2, sc2, mx2, E, 1, HID);
    edge_exp<<<cdiv(E, 256), 256, 0, stream>>>(sc2, mx2, dn2, dst, E, 1);
    recip_eps<<<cdiv(N, 256), 256, 0, stream>>>(dn2, (size_t)N);
    edge_agg<<<cdiv((long long)E * C2, 256), 256, 0, stream>>>(
        sc2, dn2, xl2, src, dst, h2, E, 1, HID, C2);
    // ---- BN2 + GELU ----
    bn_stats<<<C2, 256, 0, stream>>>(h2, N, C2, mu2, rs2);
    bn_gelu<<<cdiv((long long)N * C2, 256), 256, 0, stream>>>(
        h2, mu2, rs2, bn2g, bn2b, (size_t)N * C2, C2);

    // ---- classifier + log_softmax ----
    classify_logsoftmax<<<N, 64, 0, stream>>>(h2, Wc, bc, out, N);
}